// RDNM_54417235641042
// MI455X (gfx1250) — compile-verified
//
#include <hip/hip_runtime.h>
#include <hip/hip_bf16.h>
#include <math.h>

// Problem constants (match reference)
#define SS 10        // num_synapse
#define DD 32        // num_dendrite
#define LL 4         // num_layers
#define BB 1024      // batch
#define MM 1023      // mask rows (2^S - 1)

typedef __attribute__((ext_vector_type(2))) float v2f;
typedef __attribute__((ext_vector_type(8))) float v8f;

// ---------------------------------------------------------------------------
// Stage 1: fused  log2(sigmoid) -> (rows x 10) @ (10 x 1023 binary mask) GEMM
//          -> exp2 -> weighted reduction over mask columns.
//
// One wave handles a 16-row tile of the flattened (B*L*D = 131072) rows; 16
// consecutive rows share one layer l (D = 32).  K = 10 padded to 12 -> three
// V_WMMA_F32_16X16X4_F32 per 16-column tile, 64 column tiles per wave.
//
// The binary-mask B-tiles are identical for every wave in the kernel, so they
// are generated once per block into LDS (conflict-free lane-major v2f layout)
// and re-fetched with ds_load_b64 (imm offsets 0 / 16384 / 32768).  Working
// in log2 space makes the activation a bare v_exp_f32 (no log2e rescale).
// ---------------------------------------------------------------------------
__global__ __launch_bounds__(256) void dend_stage1(
    const float* __restrict__ x,        // (B,1,S)
    const float* __restrict__ kparm,    // (L,D,S)
    const float* __restrict__ w,        // (L,D,S)
    const float* __restrict__ q,        // (L,D,S)
    const float* __restrict__ dend_w,   // (L,M)
    const float* __restrict__ dend_b,   // (L,)
    float* __restrict__ dend_out)       // (B,L,D) scratch
{
    // B-operand tiles: [t][nt][lane] -> v2f, 48 KB, lane stride = 2 dwords
    __shared__ v2f   ldsB[3][64][32];
    // dend_w rows for all 4 layers (padded col 1023 = 0), 16 KB
    __shared__ float wsh[LL][1024];

    const int tid  = threadIdx.x;
    const int lane = tid & 31;
    const int wave = tid >> 5;

    // ---- one-time per-block fill of mask B-tiles --------------------------
    // B 4x16 f32 layout mirrors A: lanes 0-15 hold K rows kbase+0/+1,
    // lanes 16-31 hold K rows kbase+2/+3.  B[k][col] = bit(9-k) of col value.
    {
        v2f* flat = &ldsB[0][0][0];
        for (int p = tid; p < 3 * 64 * 32; p += 256) {
            const int t  = p >> 11;          // p / (64*32)
            const int nt = (p >> 5) & 63;
            const int ln = p & 31;
            const unsigned v = (unsigned)(nt * 16 + (ln & 15) + 1); // 1..1024
            const int kg0 = t * 4 + (ln >> 4) * 2;
            v2f bm;
#pragma unroll
            for (int j = 0; j < 2; ++j) {
                const int kg = kg0 + j;
                bm[j] = (kg < SS) ? (float)((v >> (SS - 1 - kg)) & 1u) : 0.0f;
            }
            flat[p] = bm;
        }
        for (int i = tid; i < LL * 1024; i += 256) {
            const int l = i >> 10, c = i & 1023;
            wsh[l][c] = (c < MM) ? dend_w[l * MM + c] : 0.0f;
        }
    }
    __syncthreads();

    const int tile = blockIdx.x * 8 + wave;    // 8192 tiles total
    const int row0 = tile * 16;
    const int l    = (row0 / DD) % LL;         // layer, constant across tile
    const int half = lane >> 4;                // 0: lanes 0-15, 1: lanes 16-31
    const int rlo  = lane & 15;                // row-in-tile (A), col-in-tile (B/C)

    // ---- per-lane A elements: log2(sigmoid(k*(w*x - q))) ------------------
    // A 16x4 f32 layout: lanes 0-15 hold K=kbase+0/+1, lanes 16-31 K=kbase+2/+3
    const int row = row0 + rlo;
    const int b   = row / (LL * DD);
    const int d   = row % DD;
    const float* kp = kparm + (size_t)(l * DD + d) * SS;
    const float* wp = w     + (size_t)(l * DD + d) * SS;
    const float* qp = q     + (size_t)(l * DD + d) * SS;
    const float* xp = x     + (size_t)b * SS;

    v2f A[3];
#pragma unroll
    for (int t = 0; t < 3; ++t) {
        const int kg0 = t * 4 + half * 2;
#pragma unroll
        for (int j = 0; j < 2; ++j) {
            const int kg = kg0 + j;
            float ls = 0.0f;
            if (kg < SS) {
                const float z  = kp[kg] * (wp[kg] * xp[kg] - qp[kg]);
                const float ez = __builtin_amdgcn_exp2f(-z * 1.44269504088896341f);
                ls = -__builtin_amdgcn_logf(1.0f + ez);   // log2(sigmoid(z))
            }
            A[t][j] = ls;
        }
    }

    // ---- main loop over 64 column tiles -----------------------------------
    v8f acc = {0.f, 0.f, 0.f, 0.f, 0.f, 0.f, 0.f, 0.f};

    for (int nt = 0; nt < 64; ++nt) {
        const v2f b0 = ldsB[0][nt][lane];
        const v2f b1 = ldsB[1][nt][lane];
        const v2f b2 = ldsB[2][nt][lane];

        v8f c = {0.f, 0.f, 0.f, 0.f, 0.f, 0.f, 0.f, 0.f};
        c = __builtin_amdgcn_wmma_f32_16x16x4_f32(false, A[0], false, b0,
                                                  (short)0, c, false, false);
        c = __builtin_amdgcn_wmma_f32_16x16x4_f32(false, A[1], false, b1,
                                                  (short)0, c, false, false);
        c = __builtin_amdgcn_wmma_f32_16x16x4_f32(false, A[2], false, b2,
                                                  (short)0, c, false, false);

        // feat = 2^c ; weighted accumulate.  C layout: VGPR i holds row
        // i + half*8, column rlo -> same weight for both lane halves.
        const float wgt = wsh[l][nt * 16 + rlo];
#pragma unroll
        for (int i = 0; i < 8; ++i)
            acc[i] = fmaf(__builtin_amdgcn_exp2f(c[i]), wgt, acc[i]);
    }

    // ---- reduce each row's partial sums across its 16-lane group ----------
#pragma unroll
    for (int m = 1; m <= 8; m <<= 1) {
#pragma unroll
        for (int i = 0; i < 8; ++i)
            acc[i] += __shfl_xor(acc[i], m, 32);
    }

    // lanes 0 and 16 hold complete sums for rows 0-7 / 8-15 of the tile
    if (rlo == 0) {
        const float bl = dend_b[l];
#pragma unroll
        for (int i = 0; i < 8; ++i)
            dend_out[row0 + half * 8 + i] = acc[i] + bl;
    }
}

// ---------------------------------------------------------------------------
// Stage 2: tiny per-batch linears:  (B,L,D) -> (B,L) -> (B,) -> sigmoid
// ---------------------------------------------------------------------------
__global__ __launch_bounds__(256) void final_stage2(
    const float* __restrict__ dend,     // (B,L,D)
    const float* __restrict__ lin_w,    // (L,D)
    const float* __restrict__ lin_b,    // (L,)
    const float* __restrict__ final_w,  // (L,)
    const float* __restrict__ final_b,  // (1,)
    const float* __restrict__ ks,       // (1,)
    float* __restrict__ out)            // (B,)
{
    const int b = blockIdx.x * blockDim.x + threadIdx.x;
    if (b >= BB) return;

    float fin = final_b[0];
#pragma unroll
    for (int l = 0; l < LL; ++l) {
        float o = lin_b[l];
        const float* dp = dend + (size_t)(b * LL + l) * DD;
#pragma unroll
        for (int d = 0; d < DD; ++d)
            o = fmaf(dp[d], lin_w[l * DD + d], o);
        fin = fmaf(o, final_w[l], fin);
    }
    out[b] = 1.0f / (1.0f + __expf(-ks[0] * fin));
}

// ---------------------------------------------------------------------------
extern "C" void kernel_launch(void* const* d_in, const int* in_sizes, int n_in,
                              void* d_out, int out_size, void* d_ws, size_t ws_size,
                              hipStream_t stream) {
    const float* x       = (const float*)d_in[0];   // (B,1,S)
    const float* kparm   = (const float*)d_in[1];   // (L,D,S)
    const float* w       = (const float*)d_in[2];   // (L,D,S)
    const float* q       = (const float*)d_in[3];   // (L,D,S)
    const float* dend_w  = (const float*)d_in[4];   // (L,M)
    const float* dend_b  = (const float*)d_in[5];   // (L,)
    const float* lin_w   = (const float*)d_in[6];   // (L,D)
    const float* lin_b   = (const float*)d_in[7];   // (L,)
    const float* final_w = (const float*)d_in[8];   // (L,)
    const float* final_b = (const float*)d_in[9];   // (1,)
    const float* ks      = (const float*)d_in[10];  // (1,)
    float* out = (float*)d_out;

    float* dend = (float*)d_ws;                     // (B,L,D) = 512 KB scratch

    // 8192 row-tiles, 8 waves (256 threads) per block -> 1024 blocks
    dend_stage1<<<1024, 256, 0, stream>>>(x, kparm, w, q, dend_w, dend_b, dend);
    final_stage2<<<(BB + 255) / 256, 256, 0, stream>>>(dend, lin_w, lin_b,
                                                       final_w, final_b, ks, out);
}